// FMNet_34093450396075
// MI455X (gfx1250) — compile-verified
//
#include <hip/hip_runtime.h>
#include <hip/hip_bf16.h>

#define GLOBAL_AS __attribute__((address_space(1)))
#define LDS_AS    __attribute__((address_space(3)))

typedef float v4f __attribute__((ext_vector_type(4)));

// out[b, h*8+c1, w*8+c2] = sigmoid(FV[b, 64 + c1*8 + c2, h, w])
// One block per (b, h): stages the 64ch x 64w tile (16KB) into LDS in
// output-linear order via async global->LDS loads (the stride-8 scatter is
// absorbed by the data mover's per-lane LDS address; global reads stay
// 128B-coalesced), then streams coalesced b128 non-temporal stores.
__global__ __launch_bounds__(256) void pixelshuffle8_sigmoid(
    const float* __restrict__ FV, float* __restrict__ out) {
  __shared__ __align__(16) float tile[4096];  // 16 KB, output-linear order

  const int bid = blockIdx.x;
  const int b   = bid >> 6;   // batch
  const int h   = bid & 63;   // input row
  const int t   = threadIdx.x;
  const int w   = t & 63;     // lane-consecutive w -> coalesced global reads
  const int cg  = t >> 6;     // channel group 0..3 (16 channels each)

  // base of FV[b, 64, h, 0]; channel stride = 64*64 = 4096 floats
  const float* gin = FV + (size_t)(b * 128 + 64) * 4096 + (size_t)h * 64;

#if __has_builtin(__builtin_amdgcn_global_load_async_to_lds_b32)
#pragma unroll
  for (int i = 0; i < 16; ++i) {
    const int c = cg * 16 + i;                           // channel 0..63
    const int p = ((c >> 3) << 9) + (w << 3) + (c & 7);  // c1*512 + w*8 + c2
    __builtin_amdgcn_global_load_async_to_lds_b32(
        (GLOBAL_AS int*)(gin + (size_t)c * 4096 + w),
        (LDS_AS int*)&tile[p],
        /*offset=*/0, /*cpol=*/0);
  }
#if __has_builtin(__builtin_amdgcn_s_wait_asynccnt)
  __builtin_amdgcn_s_wait_asynccnt(0);
#else
  asm volatile("s_wait_asynccnt 0" ::: "memory");
#endif
#else
  // Fallback: plain coalesced load + LDS scatter.
#pragma unroll
  for (int i = 0; i < 16; ++i) {
    const int c = cg * 16 + i;
    const int p = ((c >> 3) << 9) + (w << 3) + (c & 7);
    tile[p] = gin[(size_t)c * 4096 + w];
  }
#endif
  __syncthreads();

  // 8 consecutive output rows start at (b*512 + h*8)*512; tile[] is already
  // output-linear, so just stream it out with sigmoid applied.
  float* gout = out + (size_t)(b * 512 + h * 8) * 512;
  const v4f* t4 = (const v4f*)tile;
  v4f* o4 = (v4f*)gout;
#pragma unroll
  for (int q = 0; q < 4; ++q) {
    const int idx = q * 256 + t;  // lane-consecutive float4s: conflict-free
                                  // ds_load_b128 + 512B/wave coalesced stores
    v4f v = t4[idx];
    v.x = 1.0f / (1.0f + __expf(-v.x));
    v.y = 1.0f / (1.0f + __expf(-v.y));
    v.z = 1.0f / (1.0f + __expf(-v.z));
    v.w = 1.0f / (1.0f + __expf(-v.w));
    __builtin_nontemporal_store(v, o4 + idx);  // write-once: NT keeps L2 clean
  }
}

extern "C" void kernel_launch(void* const* d_in, const int* in_sizes, int n_in,
                              void* d_out, int out_size, void* d_ws, size_t ws_size,
                              hipStream_t stream) {
  const float* FV = (const float*)d_in[0];
  float* out = (float*)d_out;

  // B derived from input size: FV is (B, 128, 64, 64) f32
  const int B = in_sizes[0] / (128 * 64 * 64);

  dim3 grid(B * 64);  // one block per (b, h)
  dim3 block(256);    // 8 waves of 32
  hipLaunchKernelGGL(pixelshuffle8_sigmoid, grid, block, 0, stream, FV, out);
}